// GraphAttnGATAggregationModule_28295244546277
// MI455X (gfx1250) — compile-verified
//
#include <hip/hip_runtime.h>
#include <math.h>

// ---------------------------------------------------------------------------
// GAT aggregation for MI455X (gfx1250, wave32).
// Phase 1: score GEMM via V_WMMA_F32_16X16X4_F32 (full f32 precision),
//          bias folded into the accumulator init.
// Phase 2/3: edge softmax + aggregation with L2-resident atomics.
// ---------------------------------------------------------------------------

typedef __attribute__((ext_vector_type(2))) float v2f;
typedef __attribute__((ext_vector_type(8))) float v8f;

#define D_FEAT 128
#define NHEAD 8
#define NEG_SLOPE 0.2f

// Monotonic f32 <-> orderable u32 mapping (handles negatives) for atomicMax.
__device__ __forceinline__ unsigned f2ord(float f) {
  unsigned u = __float_as_uint(f);
  return (u & 0x80000000u) ? ~u : (u | 0x80000000u);
}
__device__ __forceinline__ float ord2f(unsigned u) {
  return (u & 0x80000000u) ? __uint_as_float(u & 0x7FFFFFFFu)
                           : __uint_as_float(~u);
}

// ---------------------------------------------------------------------------
// Init: out[:, :128] = x, out[:, 128:384] = 0; mbuf = ord(-inf)=0; sbuf = 0.
// ---------------------------------------------------------------------------
__global__ void init_kernel(const float* __restrict__ x, float* __restrict__ out,
                            unsigned* __restrict__ mbuf, float* __restrict__ sbuf,
                            int N) {
  int stride = gridDim.x * blockDim.x;
  int tid = blockIdx.x * blockDim.x + threadIdx.x;
  int totalOut = N * 384;
  for (int i = tid; i < totalOut; i += stride) {
    int n = i / 384;
    int c = i - n * 384;
    out[i] = (c < D_FEAT) ? x[n * D_FEAT + c] : 0.0f;
  }
  int totalM = N * 16;  // 2 types * 8 heads
  for (int i = tid; i < totalM; i += stride) {
    mbuf[i] = 0u;
    sbuf[i] = 0.0f;
  }
}

// ---------------------------------------------------------------------------
// Score GEMM: one wave computes a 16-row tile of scores_u (16 cols, bias in
// accumulator) and scores_v (16 cols) with V_WMMA_F32_16X16X4_F32.
// A 16x4 f32 layout: lanes 0-15 -> M=0..15, K={k0,k0+1}; lanes 16-31 K={k0+2,k0+3}.
// B 4x16: n = lane&15, same K split. C/D: vgpr r -> M = r + 8*(lane>=16), N=lane&15.
// ---------------------------------------------------------------------------
__global__ void scores_wmma_kernel(const float* __restrict__ x,
                                   const float* __restrict__ w_u,
                                   const float* __restrict__ b_u,
                                   const float* __restrict__ w_v,
                                   float* __restrict__ su,
                                   float* __restrict__ sv, int N) {
  int wave = blockIdx.x * (blockDim.x >> 5) + (threadIdx.x >> 5);
  int lane = threadIdx.x & 31;
  int row0 = wave * 16;
  if (row0 >= N) return;  // wave-uniform: EXEC all-ones for surviving waves

  int mrow = lane & 15;
  int khalf = (lane >> 4) << 1;  // 0 or 2
  int rowA = row0 + mrow;
  if (rowA >= N) rowA = N - 1;  // clamp (garbage rows computed, never stored)

  const float* xr = x + (size_t)rowA * D_FEAT;
  const float* wur = w_u + (size_t)mrow * D_FEAT;  // B row n = lane&15
  const float* wvr = w_v + (size_t)mrow * D_FEAT;

  // Bias depends only on the output column (= lane&15 for every C vgpr),
  // so seed the whole accumulator with it: D = A*B + C starts at bias.
  float bias = b_u[mrow];
  v8f accu = {bias, bias, bias, bias, bias, bias, bias, bias};
  v8f accv = {};
#pragma unroll 4
  for (int k0 = 0; k0 < D_FEAT; k0 += 4) {
    int ka = k0 + khalf;
    v2f a = *(const v2f*)(xr + ka);
    v2f bu = *(const v2f*)(wur + ka);
    v2f bv = *(const v2f*)(wvr + ka);
    accu = __builtin_amdgcn_wmma_f32_16x16x4_f32(false, a, false, bu,
                                                 (short)0, accu, false, false);
    accv = __builtin_amdgcn_wmma_f32_16x16x4_f32(false, a, false, bv,
                                                 (short)0, accv, false, false);
  }

  int nc = lane & 15;
  int mhi = (lane >> 4) * 8;
  if (row0 + 16 <= N) {
    // Full tile (all but possibly the last): branchless stores.
#pragma unroll
    for (int r = 0; r < 8; ++r) {
      int row = row0 + r + mhi;
      su[row * 16 + nc] = accu[r];
      sv[row * 16 + nc] = accv[r];
    }
  } else {
#pragma unroll
    for (int r = 0; r < 8; ++r) {
      int row = row0 + r + mhi;
      if (row < N) {
        su[row * 16 + nc] = accu[r];
        sv[row * 16 + nc] = accv[r];
      }
    }
  }
}

// ---------------------------------------------------------------------------
// Pass 1: per (edge, head) segment-max of leaky-relu score into mbuf[dst].
// ---------------------------------------------------------------------------
__global__ void edge_max_kernel(const int* __restrict__ src,
                                const int* __restrict__ dst,
                                const float* __restrict__ su,
                                const float* __restrict__ sv,
                                unsigned* __restrict__ mbuf, int E, int toff) {
  int i = blockIdx.x * blockDim.x + threadIdx.x;
  if (i >= E * NHEAD) return;
  int e = i >> 3, h = i & 7;
  int s = src[e], d = dst[e];
  float sc = su[s * 16 + toff + h] + sv[d * 16 + toff + h];
  sc = (sc >= 0.0f) ? sc : NEG_SLOPE * sc;
  atomicMax(&mbuf[d * NHEAD + h], f2ord(sc));
}

// ---------------------------------------------------------------------------
// Pass 2: per (edge, head) segment-sum of exp(score - max) into sbuf[dst].
// ---------------------------------------------------------------------------
__global__ void edge_sum_kernel(const int* __restrict__ src,
                                const int* __restrict__ dst,
                                const float* __restrict__ su,
                                const float* __restrict__ sv,
                                const unsigned* __restrict__ mbuf,
                                float* __restrict__ sbuf, int E, int toff) {
  int i = blockIdx.x * blockDim.x + threadIdx.x;
  if (i >= E * NHEAD) return;
  int e = i >> 3, h = i & 7;
  int s = src[e], d = dst[e];
  float sc = su[s * 16 + toff + h] + sv[d * 16 + toff + h];
  sc = (sc >= 0.0f) ? sc : NEG_SLOPE * sc;
  float m = ord2f(mbuf[d * NHEAD + h]);
  atomicAdd(&sbuf[d * NHEAD + h], __expf(sc - m));
}

// ---------------------------------------------------------------------------
// Pass 3: one wave per edge. Lanes 0-7 compute p[h]; shfl-broadcast; each
// lane gathers a float4 of x[src] and scatters 4 f32 atomics into out[dst].
// Channel c maps to head h = c & 7 (xh[n,d,h] = x[n, d*8+h]).
// ---------------------------------------------------------------------------
__global__ void edge_agg_kernel(const int* __restrict__ src,
                                const int* __restrict__ dst,
                                const float* __restrict__ x,
                                const float* __restrict__ su,
                                const float* __restrict__ sv,
                                const unsigned* __restrict__ mbuf,
                                const float* __restrict__ sbuf,
                                float* __restrict__ out, int E, int toff,
                                int coff) {
  int wave = blockIdx.x * (blockDim.x >> 5) + (threadIdx.x >> 5);
  int lane = threadIdx.x & 31;
  if (wave >= E) return;
  int s = src[wave], d = dst[wave];

  float p = 0.0f;
  if (lane < NHEAD) {
    float sc = su[s * 16 + toff + lane] + sv[d * 16 + toff + lane];
    sc = (sc >= 0.0f) ? sc : NEG_SLOPE * sc;
    float m = ord2f(mbuf[d * NHEAD + lane]);
    float denom = sbuf[d * NHEAD + lane];
    p = __expf(sc - m) / denom;
  }

  // Channels for this lane: c = lane*4 + j  ->  h = (lane*4 + j) & 7.
  int hbase = (lane & 1) * 4;
  float p0 = __shfl(p, hbase + 0);
  float p1 = __shfl(p, hbase + 1);
  float p2 = __shfl(p, hbase + 2);
  float p3 = __shfl(p, hbase + 3);

  float4 xv = *(const float4*)(x + (size_t)s * D_FEAT + lane * 4);
  float* ob = out + (size_t)d * 384 + coff + lane * 4;
  atomicAdd(ob + 0, xv.x * p0);
  atomicAdd(ob + 1, xv.y * p1);
  atomicAdd(ob + 2, xv.z * p2);
  atomicAdd(ob + 3, xv.w * p3);
}

// ---------------------------------------------------------------------------
extern "C" void kernel_launch(void* const* d_in, const int* in_sizes, int n_in,
                              void* d_out, int out_size, void* d_ws,
                              size_t ws_size, hipStream_t stream) {
  const float* x = (const float*)d_in[0];
  const float* w_u = (const float*)d_in[1];
  const float* b_u = (const float*)d_in[2];
  const float* w_v = (const float*)d_in[3];
  const int* src0 = (const int*)d_in[4];
  const int* dst0 = (const int*)d_in[5];
  const int* src1 = (const int*)d_in[6];
  const int* dst1 = (const int*)d_in[7];

  int N = in_sizes[0] / D_FEAT;
  int E = in_sizes[4];
  float* out = (float*)d_out;

  // Workspace layout (all L2-resident, ~12.8 MB at N=50000):
  //   su[N*16], sv[N*16], mbuf[2][N*8] (u32), sbuf[2][N*8]
  float* su = (float*)d_ws;
  float* sv = su + (size_t)N * 16;
  unsigned* mbuf = (unsigned*)(sv + (size_t)N * 16);
  float* sbuf = (float*)(mbuf + (size_t)N * 16);

  // 1) init output (x copy + zeroed agg region) and softmax stats.
  init_kernel<<<4096, 256, 0, stream>>>(x, out, mbuf, sbuf, N);

  // 2) score GEMM with WMMA: one wave per 16-row tile, 8 waves per block.
  int tiles = (N + 15) / 16;
  int sblocks = (tiles + 7) / 8;
  scores_wmma_kernel<<<sblocks, 256, 0, stream>>>(x, w_u, b_u, w_v, su, sv, N);

  // 3) two edge types, three passes each.
  int ehblocks = (E * NHEAD + 255) / 256;  // thread per (edge, head)
  int eablocks = (E + 7) / 8;              // wave per edge, 8 per block

  unsigned* m0 = mbuf;
  unsigned* m1 = mbuf + (size_t)N * NHEAD;
  float* s0 = sbuf;
  float* s1 = sbuf + (size_t)N * NHEAD;

  edge_max_kernel<<<ehblocks, 256, 0, stream>>>(src0, dst0, su, sv, m0, E, 0);
  edge_sum_kernel<<<ehblocks, 256, 0, stream>>>(src0, dst0, su, sv, m0, s0, E, 0);
  edge_agg_kernel<<<eablocks, 256, 0, stream>>>(src0, dst0, x, su, sv, m0, s0,
                                                out, E, 0, 128);

  edge_max_kernel<<<ehblocks, 256, 0, stream>>>(src1, dst1, su, sv, m1, E, 8);
  edge_sum_kernel<<<ehblocks, 256, 0, stream>>>(src1, dst1, su, sv, m1, s1, E, 8);
  edge_agg_kernel<<<eablocks, 256, 0, stream>>>(src1, dst1, x, su, sv, m1, s1,
                                                out, E, 8, 256);
}